// DAN_7164005450416
// MI455X (gfx1250) — compile-verified
//
#include <hip/hip_runtime.h>
#include <hip/hip_bf16.h>
#include <math.h>

// ---------------------------------------------------------------------------
// DAN classifier for MI455X (gfx1250, wave32).
//   avg   = sum_s emb[ids*mask] / count(drop_mask)                [256,300]
//   h1    = relu(avg @ W1^T + b1)                                 [256,300]
//   h2    = relu(h1  @ W2^T + b2)                                 [256,300]
//   logit = h2 @ W3^T + b3                                        [256,7]
//   loss  = mean NLL(log_softmax(logit), labels)
// GEMMs: V_WMMA_F32_16X16X4_F32 (f32 in / f32 acc, K-step 4; 300 = 75*4).
// Weights/biases are zero-padded in N to a multiple of 16 so the WMMA K-loop
// is branch-free and EXEC stays all-ones (no saveexec around loads).
// ---------------------------------------------------------------------------

typedef float v2f __attribute__((ext_vector_type(2)));
typedef float v8f __attribute__((ext_vector_type(8)));

#define Bsz     256
#define Ssz     512
#define Hdim    300
#define Hpad    304          // 19 * 16
#define NLAB    7
#define NLABpad 16
#define PFDIST  4            // prefetch distance (rows of S) in gather loop

// ---------------------------------------------------------------------------
// Zero-padding copy: dst[i] = (i < nvalid) ? src[i] : 0   for i < total.
// Works for weights ([Nvalid,K] -> [Npad,K], same linear layout) and biases.
// ---------------------------------------------------------------------------
__global__ void __launch_bounds__(256)
pad_copy(const float* __restrict__ src, float* __restrict__ dst,
         int nvalid, int total)
{
    const int i = blockIdx.x * blockDim.x + threadIdx.x;
    if (i < total) dst[i] = (i < nvalid) ? src[i] : 0.0f;
}

// ---------------------------------------------------------------------------
// Kernel 1: masked embedding gather + sum over S, divide by count(drop_mask).
// One block per batch row, 320 threads (10 waves). Writes avg padded to Hpad.
// ---------------------------------------------------------------------------
__global__ void __launch_bounds__(320)
dan_gather_avg(const int* __restrict__ ids,
               const int* __restrict__ amask,
               const int* __restrict__ drop,
               const float* __restrict__ emb,
               float* __restrict__ avg)
{
    __shared__ int s_id[Ssz];
    __shared__ int s_cnt;

    const int b = blockIdx.x;
    const int t = threadIdx.x;

    if (t == 0) s_cnt = 0;
    __syncthreads();

    int cnt = 0;
    for (int s = t; s < Ssz; s += 320) {
        const int base = b * Ssz + s;
        s_id[s] = ids[base] * amask[base];
        cnt += drop[base];
    }
    if (cnt) atomicAdd(&s_cnt, cnt);
    __syncthreads();

    const float inv_nz = 1.0f / (float)s_cnt;   // matches reference (may be inf)

    if (t < Hpad) {
        float acc = 0.0f;
        if (t < Hdim) {
            #pragma unroll 2
            for (int s = 0; s < Ssz; ++s) {
                if (s + PFDIST < Ssz) {
                    const size_t pidx = (size_t)s_id[s + PFDIST] * Hdim + t;
                    __builtin_prefetch(&emb[pidx], 0, 0);   // global_prefetch_b8
                }
                acc += emb[(size_t)s_id[s] * Hdim + t];
            }
            acc *= inv_nz;
        }
        avg[(size_t)b * Hpad + t] = (t < Hdim) ? acc : 0.0f;  // zero pad cols
    }
}

// ---------------------------------------------------------------------------
// Kernel 2: Out = act(A @ Wp^T + biasp) via V_WMMA_F32_16X16X4_F32.
// One wave per 16x16 output tile. A is [256, Hpad] (pad cols zero),
// Wp is [Npad, Hdim] row-major zero-padded, biasp is [Npad] zero-padded.
//
// VGPR layouts (ISA 7.12.2, f32 16x16x4):
//   A 16x4 : lanes 0-15 rows M with K{k,k+1}; lanes 16-31 K{k+2,k+3}
//   B 4x16 : lanes 0-15 cols N with K{k,k+1}; lanes 16-31 K{k+2,k+3}
//   C/D    : VGPR r -> M = r (+8 for lanes 16-31), N = lane&15
// ---------------------------------------------------------------------------
template<bool RELU, bool FINAL>
__global__ void __launch_bounds__(32)
dan_gemm_wmma(const float* __restrict__ A,
              const float* __restrict__ Wp,
              const float* __restrict__ biasp,
              float* __restrict__ Out,
              int ldOut)
{
    const int tile = blockIdx.x;
    const int mt   = tile & 15;        // 16 M tiles
    const int nt   = tile >> 4;

    const int lane = threadIdx.x & 31;
    const int half = lane >> 4;        // selects K{0,1} vs K{2,3} pair
    const int l15  = lane & 15;

    const int m = mt * 16 + l15;       // A row this lane feeds
    const int n = nt * 16 + l15;       // B col / D col this lane owns

    const float* __restrict__ Arow = A  + (size_t)m * Hpad + 2 * half;
    const float* __restrict__ Wrow = Wp + (size_t)n * Hdim + 2 * half;

    v8f c = {0.f, 0.f, 0.f, 0.f, 0.f, 0.f, 0.f, 0.f};

    #pragma unroll 5                    // 75 K-steps total
    for (int k = 0; k < Hdim; k += 4) {
        const v2f a  = *(const v2f*)(Arow + k);   // global_load_b64
        const v2f bb = *(const v2f*)(Wrow + k);   // global_load_b64
        c = __builtin_amdgcn_wmma_f32_16x16x4_f32(
                /*neg_a=*/false, a, /*neg_b=*/false, bb,
                /*c_mod=*/(short)0, c, /*reuse_a=*/false, /*reuse_b=*/false);
    }

    const float bn = biasp[n];          // zero on pad columns

    #pragma unroll
    for (int r = 0; r < 8; ++r) {
        const int mr = mt * 16 + r + half * 8;   // D row for this VGPR
        float v = c[r] + bn;                     // pad cols: 0 + 0 = 0
        if (RELU) v = fmaxf(v, 0.f);
        if (FINAL) {
            if (n < NLAB) Out[(size_t)mr * ldOut + n] = v;
        } else {
            Out[(size_t)mr * ldOut + n] = v;     // pad cols self-zero
        }
    }
}

// ---------------------------------------------------------------------------
// Kernel 3: log_softmax + mean NLL over 256 rows of 7 logits.
// ---------------------------------------------------------------------------
__global__ void __launch_bounds__(Bsz)
dan_loss(const float* __restrict__ logits,
         const int* __restrict__ labels,
         float* __restrict__ loss_out)
{
    __shared__ float red[Bsz];
    const int b = threadIdx.x;

    float x[NLAB];
    #pragma unroll
    for (int i = 0; i < NLAB; ++i) x[i] = logits[b * NLAB + i];

    float mx = x[0];
    #pragma unroll
    for (int i = 1; i < NLAB; ++i) mx = fmaxf(mx, x[i]);

    float s = 0.f;
    #pragma unroll
    for (int i = 0; i < NLAB; ++i) s += __expf(x[i] - mx);
    const float lse = __logf(s) + mx;

    red[b] = lse - x[labels[b]];       // -logp[label]
    __syncthreads();

    #pragma unroll
    for (int off = Bsz / 2; off > 0; off >>= 1) {
        if (b < off) red[b] += red[b + off];
        __syncthreads();
    }
    if (b == 0) loss_out[0] = red[0] / (float)Bsz;
}

// ---------------------------------------------------------------------------
extern "C" void kernel_launch(void* const* d_in, const int* in_sizes, int n_in,
                              void* d_out, int out_size, void* d_ws, size_t ws_size,
                              hipStream_t stream)
{
    const int*   ids   = (const int*)  d_in[0];   // [256,512]
    const int*   amask = (const int*)  d_in[1];   // [256,512]
    const int*   labels= (const int*)  d_in[2];   // [256]
    const int*   drop  = (const int*)  d_in[3];   // [256,512]
    const float* emb   = (const float*)d_in[4];   // [400000,300]
    const float* W1    = (const float*)d_in[5];   // [300,300]
    const float* b1    = (const float*)d_in[6];   // [300]
    const float* W2    = (const float*)d_in[7];   // [300,300]
    const float* b2    = (const float*)d_in[8];   // [300]
    const float* W3    = (const float*)d_in[9];   // [7,300]
    const float* b3    = (const float*)d_in[10];  // [7]

    float* out = (float*)d_out;                   // [0]=loss, [1..1792]=logits

    // ---- workspace layout (floats) ----
    float* avg = (float*)d_ws;                    // [256,304]
    float* h1  = avg + (size_t)Bsz * Hpad;        // [256,304]
    float* h2  = h1  + (size_t)Bsz * Hpad;        // [256,304]
    float* W1p = h2  + (size_t)Bsz * Hpad;        // [304,300]
    float* W2p = W1p + (size_t)Hpad * Hdim;       // [304,300]
    float* W3p = W2p + (size_t)Hpad * Hdim;       // [16,300]
    float* b1p = W3p + (size_t)NLABpad * Hdim;    // [304]
    float* b2p = b1p + Hpad;                      // [304]
    float* b3p = b2p + Hpad;                      // [16]

    // ---- zero-pad weights/biases in N so WMMA loops are branch-free ----
    const int wtot = Hpad * Hdim, wval = Hdim * Hdim;          // 91200 / 90000
    pad_copy<<<(wtot + 255) / 256, 256, 0, stream>>>(W1, W1p, wval, wtot);
    pad_copy<<<(wtot + 255) / 256, 256, 0, stream>>>(W2, W2p, wval, wtot);
    const int w3tot = NLABpad * Hdim, w3val = NLAB * Hdim;     // 4800 / 2100
    pad_copy<<<(w3tot + 255) / 256, 256, 0, stream>>>(W3, W3p, w3val, w3tot);
    pad_copy<<<(Hpad + 255) / 256, 256, 0, stream>>>(b1, b1p, Hdim, Hpad);
    pad_copy<<<(Hpad + 255) / 256, 256, 0, stream>>>(b2, b2p, Hdim, Hpad);
    pad_copy<<<1, 256, 0, stream>>>(b3, b3p, NLAB, NLABpad);

    // ---- 1) embedding gather + average ----
    dan_gather_avg<<<Bsz, 320, 0, stream>>>(ids, amask, drop, emb, avg);

    const int Mtiles = Bsz / 16;                  // 16
    const int Ntiles = Hpad / 16;                 // 19

    // ---- 2) h1 = relu(avg @ W1^T + b1) ----
    dan_gemm_wmma<true, false><<<Mtiles * Ntiles, 32, 0, stream>>>(
        avg, W1p, b1p, h1, Hpad);

    // ---- 3) h2 = relu(h1 @ W2^T + b2) ----
    dan_gemm_wmma<true, false><<<Mtiles * Ntiles, 32, 0, stream>>>(
        h1, W2p, b2p, h2, Hpad);

    // ---- 4) logits = h2 @ W3^T + b3 -> d_out+1 (ld = 7), one N tile ----
    dan_gemm_wmma<false, true><<<Mtiles * 1, 32, 0, stream>>>(
        h2, W3p, b3p, out + 1, NLAB);

    // ---- 5) loss ----
    dan_loss<<<1, Bsz, 0, stream>>>(out + 1, labels, out);
}